// PositionEncoder_42374147342670
// MI455X (gfx1250) — compile-verified
//
#include <hip/hip_runtime.h>
#include <stdint.h>

// PositionEncoder for MI455X (gfx1250, wave32).
// Memory-bound concat/gather: stage emb rows 0..26 + x slab in LDS via CDNA5
// GLOBAL_LOAD_ASYNC_TO_LDS (ASYNCcnt), assemble, stream b128 non-temporal stores.

typedef __attribute__((ext_vector_type(4))) float float4v;

#define NNODES   26
#define TOK      64          // tokens per block
#define THREADS  256         // 8 wave32s
#define EMB_ROWS 27          // indices 0..26 used
#define EMBF     (EMB_ROWS * 128)   // 3456 floats, emb region in LDS
#define XBASE    EMBF               // x region base (float index)
#define SMEMF    (EMBF + TOK * 16)  // 4480 floats total

// Node coordinates: double literals cast to float == np.float32(value) path.
__device__ __constant__ float c_nx[NNODES] = {
    (float)0.5454545454545454,  (float)0.6022727272727273,  (float)0.5454545454545454,
    (float)0.6022727272727273,  (float)0.4772727272727273,  (float)0.42045454545454547,
    (float)0.42045454545454547, (float)0.4772727272727273,  (float)0.32954545454545453,
    (float)0.42045454545454547, (float)0.4772727272727273,  (float)0.4772727272727273,
    (float)0.42045454545454547, (float)0.32954545454545453, (float)0.5727272727272728,
    (float)0.7613636363636364,  (float)0.8181818181818182,  (float)0.8181818181818182,
    (float)0.7613636363636364,  (float)0.7909090909090909,  (float)0.9431818181818182,
    (float)1.0,                 (float)1.0,                 (float)0.9431818181818182,
    (float)0.9727272727272728,  (float)0.9727272727272728
};
__device__ __constant__ float c_ny[NNODES] = {
    (float)0.76,  (float)0.76,  (float)0.86,  (float)0.86,  (float)0.76,
    (float)0.76,  (float)0.86,  (float)0.86,  (float)0.808, (float)0.48,
    (float)0.48,  (float)0.38,  (float)0.38,  (float)0.428, (float)0.62,
    (float)0.76,  (float)0.76,  (float)0.86,  (float)0.86,  (float)0.62,
    (float)0.76,  (float)0.76,  (float)0.86,  (float)0.86,  (float)0.62,
    (float)1.0
};

__device__ __forceinline__ void async_copy16(uint32_t lds_off, uint64_t gaddr) {
    // CDNA5 async DMA: copies 16B per active lane from global address gaddr to LDS address lds_off (ASYNCcnt).
    asm volatile("global_load_async_to_lds_b128 %0, %1, off"
                 :: "v"(lds_off), "v"(gaddr)
                 : "memory");
}

__global__ __launch_bounds__(THREADS)
void PositionEncoder_kernel(const float* __restrict__ x,
                            const float* __restrict__ emb,
                            float* __restrict__ out,
                            int nTok)
{
    __shared__ __align__(16) float smem[SMEMF];
    __shared__ int sidx[2 * TOK];

    const int tid    = threadIdx.x;
    const int token0 = blockIdx.x * TOK;
    const int tokHere = (nTok - token0 < TOK) ? (nTok - token0) : TOK;
    if (tokHere <= 0) return;

    // ---------------- Phase 0: async global -> LDS staging ----------------
    // emb rows 0..26: 27*128 floats = 864 float4, contiguous from emb base.
    for (int i = tid; i < EMBF / 4; i += THREADS) {
        uint32_t loff = (uint32_t)(uintptr_t)&smem[i * 4];
        async_copy16(loff, (uint64_t)(uintptr_t)(emb + (size_t)i * 4));
    }
    // x slab for this block: tokHere*16 floats = tokHere*4 float4 (<= 256).
    {
        const int n4 = tokHere * 4;
        if (tid < n4) {
            uint32_t loff = (uint32_t)(uintptr_t)&smem[XBASE + tid * 4];
            async_copy16(loff, (uint64_t)(uintptr_t)(x + (size_t)token0 * 16 + (size_t)tid * 4));
        }
    }
    asm volatile("s_wait_asynccnt 0" ::: "memory");
    __syncthreads();

    // ---------------- Phase 1: per-token node matching ----------------
    if (tid < tokHere) {
        const float px1 = smem[XBASE + tid * 16 + 4];
        const float py1 = smem[XBASE + tid * 16 + 5];
        const float px2 = smem[XBASE + tid * 16 + 8];
        const float py2 = smem[XBASE + tid * 16 + 9];
        int i1 = 0, i2 = 0;
        // Descending overwrite == argmax-of-first-True (+1), 0 if none.
        #pragma unroll
        for (int i = NNODES - 1; i >= 0; --i) {
            const float nx = c_nx[i], ny = c_ny[i];
            const float tx = 0.01f + 1e-5f * fabsf(nx);
            const float ty = 0.01f + 1e-5f * fabsf(ny);
            const bool c1 = (fabsf(px1 - nx) <= tx) && (fabsf(py1 - ny) <= ty);
            const bool c2 = (fabsf(px2 - nx) <= tx) && (fabsf(py2 - ny) <= ty);
            i1 = c1 ? (i + 1) : i1;
            i2 = c2 ? (i + 1) : i2;
        }
        sidx[tid]       = i1;
        sidx[TOK + tid] = i2;
    }
    __syncthreads();

    // ---------------- Phase 2: assemble + stream out ----------------
    // Output layout per token: [x0:6 | emb[i1] | x6:10 | emb[i2] | x10:16] = 272 f32
    // = 68 float4. v enumerates float4s contiguously across tokens in this block,
    // so stores are perfectly coalesced 16B/lane streams.
    float4v* __restrict__ outv = (float4v*)out;
    const size_t obase = (size_t)token0 * 68;
    const int vmax = tokHere * 68;
    for (int v = tid; v < vmax; v += THREADS) {
        const int token = v / 68;
        const int j     = v - token * 68;
        const int i1 = sidx[token];
        const int i2 = sidx[TOK + token];
        const int xb = XBASE + token * 16;
        float4v r;
        #pragma unroll
        for (int c = 0; c < 4; ++c) {
            const int colc = 4 * j + c;
            int src;
            if (colc < 6)        src = xb + colc;                  // x[0:6]
            else if (colc < 134) src = i1 * 128 + (colc - 6);      // emb[i1]
            else if (colc < 138) src = xb + (colc - 128);          // x[6:10]
            else if (colc < 266) src = i2 * 128 + (colc - 138);    // emb[i2]
            else                 src = xb + (colc - 256);          // x[10:16]
            r[c] = smem[src];
        }
        __builtin_nontemporal_store(r, &outv[obase + v]);
    }
}

extern "C" void kernel_launch(void* const* d_in, const int* in_sizes, int n_in,
                              void* d_out, int out_size, void* d_ws, size_t ws_size,
                              hipStream_t stream) {
    (void)n_in; (void)out_size; (void)d_ws; (void)ws_size;
    const float* x   = (const float*)d_in[0];   // [128,2048,16] f32
    const float* emb = (const float*)d_in[1];   // [100,128] f32
    float* out = (float*)d_out;                 // [128,2048,272] f32

    const int nTok = in_sizes[0] / 16;          // 262144
    const int blocks = (nTok + TOK - 1) / TOK;  // 4096
    PositionEncoder_kernel<<<blocks, THREADS, 0, stream>>>(x, emb, out, nTok);
}